// Hyper_Attention_Class_88321707475499
// MI455X (gfx1250) — compile-verified
//
#include <hip/hip_runtime.h>

// ---------------- problem constants (from reference) ----------------
#define NN   100000   // nodes
#define MM   100000   // hyperedges
#define EE   1600000  // incidences
#define FIN  500
#define H    8
#define O1   8
#define C1   64       // H*O1
#define O2   8
#define NEG  0.2f

typedef __attribute__((ext_vector_type(2))) float        v2f;
typedef __attribute__((ext_vector_type(8))) float        v8f;
typedef __attribute__((ext_vector_type(4))) unsigned int v4u;
typedef __attribute__((ext_vector_type(4))) int          v4i;
typedef __attribute__((ext_vector_type(8))) int          v8i;

#ifdef __has_builtin
# if __has_builtin(__builtin_amdgcn_tensor_load_to_lds) && __has_builtin(__builtin_amdgcn_s_wait_tensorcnt)
#  define HAS_TDM 1
# endif
#endif
#ifndef HAS_TDM
# define HAS_TDM 0
#endif

// ---------------- small helpers ----------------
__device__ __forceinline__ void atomicMaxF(float* addr, float val) {
  // float max via int max (positives) / uint min (negatives); init must be -inf
  if (val >= 0.0f) atomicMax((int*)addr, __float_as_int(val));
  else             atomicMin((unsigned int*)addr, __float_as_uint(val));
}

__global__ void fill_k(float* __restrict__ p, float v, long n) {
  long i = (long)blockIdx.x * blockDim.x + threadIdx.x;
  if (i < n) p[i] = v;
}

__global__ void invert_k(float* __restrict__ p, long n) {
  long i = (long)blockIdx.x * blockDim.x + threadIdx.x;
  if (i < n) { float d = p[i]; p[i] = (d > 0.0f) ? (1.0f / d) : 0.0f; }
}

// v[i] *= s[i >> shift]
__global__ void scale_rows_k(float* __restrict__ v, const float* __restrict__ s,
                             int shift, long n) {
  long i = (long)blockIdx.x * blockDim.x + threadIdx.x;
  if (i < n) v[i] *= s[i >> shift];
}

// ---------------- TDM: 2D tile load Global -> LDS (CDNA5 Tensor Data Mover) ----
#if HAS_TDM
// Builds D# per CDNA5 ISA ch.8: group0 = {count/type, lds_addr, global_addr},
// group1 = {data_size=4B, pad ctrl, tensor/tile dims, dim0 stride}. Tile dims
// equal the transfer extent so no OOB path is exercised.
__device__ __forceinline__ void tdm_load_2d(unsigned lds_off, const void* g,
                                            unsigned tile_w, unsigned tile_h,
                                            unsigned stride_elems,
                                            int pad_iv /* -1 = none */, int pad_amt) {
  unsigned long long ga = (unsigned long long)(uintptr_t)g;
  v4u g0;
  g0.x = 1u;                                               // count=1 (valid), user mode
  g0.y = lds_off;                                          // LDS byte address
  g0.z = (unsigned)ga;                                     // global_addr[31:0]
  g0.w = (unsigned)((ga >> 32) & 0x01FFFFFFull) | (2u << 30); // [56:32] + type=2
  unsigned w0 = (2u << 16);                                // data_size = 4 bytes
  if (pad_iv >= 0)
    w0 |= (1u << 20) | ((unsigned)pad_iv << 22) | ((unsigned)pad_amt << 25);
  v8i g1;
  g1[0] = (int)w0;                                         // mask=0, pad ctrl
  g1[1] = (int)((tile_w & 0xFFFFu) << 16);                 // tensor_dim0 = tile_w
  g1[2] = (int)((tile_w >> 16) | ((tile_h & 0xFFFFu) << 16)); // tensor_dim1 = tile_h
  g1[3] = (int)((tile_h >> 16) | ((tile_w & 0xFFFFu) << 16)); // tile_dim0
  g1[4] = (int)(tile_h & 0xFFFFu);                         // tile_dim1 (tile_dim2=0)
  g1[5] = (int)stride_elems;                               // tensor_dim0_stride
  g1[6] = 0;
  g1[7] = 0;
  v4i z = {};
#if __clang_major__ >= 23
  v8i z8 = {};
  __builtin_amdgcn_tensor_load_to_lds(g0, g1, z, z, z8, 0);
#else
  __builtin_amdgcn_tensor_load_to_lds(g0, g1, z, z, 0);
#endif
}

__device__ __forceinline__ void issue_chunk(const float* x, const float* W,
                                            long rowBase, int rowsHere, int ch,
                                            float* a_buf, float* b_buf) {
  const int k0 = ch * 32;
  const int kw = (ch < 15) ? 32 : 20;       // 500 = 15*32 + 20
  // A slab: rowsHere x kw from x (row stride FIN); pad rows 32->33 DW (iv code 4 = 32 DW)
  tdm_load_2d((unsigned)(uintptr_t)a_buf, x + rowBase * FIN + k0,
              (unsigned)kw, (unsigned)rowsHere, (unsigned)FIN,
              (kw == 32) ? 4 : -1, 0);
  // B slab: kw x 64 from W1 (row stride 64); pad rows 64->65 DW (iv code 5 = 64 DW)
  tdm_load_2d((unsigned)(uintptr_t)b_buf, W + (size_t)k0 * C1,
              64u, (unsigned)kw, 64u, 5, 0);
}
#endif

// inner product step: 16x64 C block, K advanced by 4 per WMMA, A/B from LDS
__device__ __forceinline__ void mma_chunk(const float* __restrict__ A,
                                          const float* __restrict__ Bm,
                                          int RS, int kw, int rl, int half, int ln,
                                          v8f& c0, v8f& c1, v8f& c2, v8f& c3) {
  for (int kk = 0; kk < kw; kk += 4) {
    v2f a;                                   // A 16x4 f32: lanes<16 K=0,1; >=16 K=2,3
    a.x = A[rl * RS + kk + 2 * half];
    a.y = A[rl * RS + kk + 2 * half + 1];
    const float* b = Bm + (kk + 2 * half) * 65 + ln;
    v2f b0 = { b[0],  b[65] };
    v2f b1 = { b[16], b[65 + 16] };
    v2f b2 = { b[32], b[65 + 32] };
    v2f b3 = { b[48], b[65 + 48] };
    c0 = __builtin_amdgcn_wmma_f32_16x16x4_f32(false, a, false, b0, (short)0, c0, false, false);
    c1 = __builtin_amdgcn_wmma_f32_16x16x4_f32(false, a, false, b1, (short)0, c1, false, false);
    c2 = __builtin_amdgcn_wmma_f32_16x16x4_f32(false, a, false, b2, (short)0, c2, false, false);
    c3 = __builtin_amdgcn_wmma_f32_16x16x4_f32(false, a, false, b3, (short)0, c3, false, false);
  }
}

// ---------------- GEMM1: xw = x[N,500] @ W1[500,64] --------------------------
// Block = 256 thr = 8 waves = 128 rows. TDM double-buffers 128x32 A-slabs and
// 32x64 B-slabs into LDS (padded row strides 33/65 vs 64 banks); waves run
// V_WMMA_F32_16X16X4_F32 out of LDS while TDM fetches the next slab.
__global__ void __launch_bounds__(256) gemm1_wmma(const float* __restrict__ x,
                                                  const float* __restrict__ W,
                                                  float* __restrict__ xw) {
  __shared__ float sA[2][128 * 33];   // 33,792 B
  __shared__ float sB[2][32 * 65];    // 16,640 B
  const int tid  = threadIdx.x;
  const int wid  = tid >> 5;
  const int lane = tid & 31;
  const int half = lane >> 4;
  const int ln   = lane & 15;
  const long rowBase  = (long)blockIdx.x * 128;
  const int  rowsHere = (NN - rowBase >= 128) ? 128 : (int)(NN - rowBase);
  const int  rl = wid * 16 + ln;      // local A row this lane feeds

  v8f c0 = {}, c1 = {}, c2 = {}, c3 = {};

#if HAS_TDM
  if (wid == 0) {
    issue_chunk(x, W, rowBase, rowsHere, 0, &sA[0][0], &sB[0][0]);
    __builtin_amdgcn_s_wait_tensorcnt(0);
  }
  __syncthreads();
  for (int ch = 0; ch < 16; ++ch) {
    const int buf = ch & 1;
    const int kw  = (ch < 15) ? 32 : 20;
    const int RS  = (ch < 15) ? 33 : 20;
    if (wid == 0 && ch + 1 < 16)
      issue_chunk(x, W, rowBase, rowsHere, ch + 1, &sA[buf ^ 1][0], &sB[buf ^ 1][0]);
    mma_chunk(&sA[buf][0], &sB[buf][0], RS, kw, rl, half, ln, c0, c1, c2, c3);
    if (wid == 0 && ch + 1 < 16)
      __builtin_amdgcn_s_wait_tensorcnt(0);   // next slab landed in LDS
    __syncthreads();
  }
#else
  for (int ch = 0; ch < 16; ++ch) {
    const int kw = (ch < 15) ? 32 : 20;
    const int RS = (ch < 15) ? 33 : 20;
    const int k0 = ch * 32;
    __syncthreads();
    for (int idx = tid; idx < rowsHere * kw; idx += 256) {
      int r = idx / kw, c = idx - r * kw;
      sA[0][r * RS + c] = x[(rowBase + r) * FIN + k0 + c];
    }
    for (int idx = tid; idx < kw * 64; idx += 256) {
      int r = idx >> 6, c = idx & 63;
      sB[0][r * 65 + c] = W[(size_t)(k0 + r) * 64 + c];
    }
    __syncthreads();
    mma_chunk(&sA[0][0], &sB[0][0], RS, kw, rl, half, ln, c0, c1, c2, c3);
  }
#endif

#pragma unroll
  for (int r = 0; r < 8; ++r) {        // C: VGPR r -> M = r (lanes<16) / r+8
    long m = rowBase + wid * 16 + r + 8 * half;
    if (m < NN) {
      float* o = xw + m * C1 + ln;
      o[0]  = c0[r];
      o[16] = c1[r];
      o[32] = c2[r];
      o[48] = c3[r];
    }
  }
}

// ---------------- GEMM2: y = h[N,64] @ W2[64,8], 8-wide padded to one 16-tile ----
__global__ void gemm2_wmma(const float* __restrict__ h, const float* __restrict__ W2,
                           float* __restrict__ y) {
  const int gwave = (int)((blockIdx.x * blockDim.x + threadIdx.x) >> 5);
  const int lane  = threadIdx.x & 31;
  const int NT    = NN / 16;
  if (gwave >= NT) return;
  const int half = lane >> 4;
  const int ln   = lane & 15;
  const float* hr = h + ((long)gwave * 16 + ln) * C1;
  v8f c = {};
  for (int k = 0; k < C1; k += 4) {
    v2f a, b;
    a.x = hr[k + 2 * half];
    a.y = hr[k + 2 * half + 1];
    b.x = (ln < O2) ? W2[(size_t)(k + 2 * half) * O2 + ln]     : 0.0f;
    b.y = (ln < O2) ? W2[(size_t)(k + 2 * half + 1) * O2 + ln] : 0.0f;
    c = __builtin_amdgcn_wmma_f32_16x16x4_f32(false, a, false, b, (short)0, c, false, false);
  }
  if (ln < O2) {
#pragma unroll
    for (int r = 0; r < 8; ++r)
      y[((long)gwave * 16 + r + 8 * half) * O2 + ln] = c[r];
  }
}

// ---------------- graph kernels (per-incidence, L2-resident operands) ----------
__global__ void count_deg_k(const long long* __restrict__ ei,
                            float* __restrict__ Dcnt, float* __restrict__ Bcnt) {
  int e = blockIdx.x * blockDim.x + threadIdx.x;
  if (e >= EE) return;
  atomicAdd(&Dcnt[(int)ei[e]], 1.0f);
  atomicAdd(&Bcnt[(int)ei[(size_t)EE + e]], 1.0f);
}

__global__ void alpha_max_k(const long long* __restrict__ ei, const float* __restrict__ xw,
                            const float* __restrict__ att, float* __restrict__ alpha,
                            float* __restrict__ amax) {
  int e = blockIdx.x * blockDim.x + threadIdx.x;
  if (e >= EE) return;
  int row = (int)ei[e];
  int col = (int)ei[(size_t)EE + e];
  const float* xi = xw + (size_t)row * C1;
  const float* xj = xw + (size_t)col * C1;
#pragma unroll
  for (int h = 0; h < H; ++h) {
    float s = 0.0f;
#pragma unroll
    for (int c = 0; c < O1; ++c)
      s += att[h * 2 * O1 + c]      * xi[h * O1 + c]
         + att[h * 2 * O1 + O1 + c] * xj[h * O1 + c];
    s = (s > 0.0f) ? s : NEG * s;                 // leaky_relu
    alpha[(size_t)e * H + h] = s;
    atomicMaxF(&amax[(size_t)row * H + h], s);
  }
}

__global__ void exp_sum_k(const long long* __restrict__ ei, float* __restrict__ alpha,
                          const float* __restrict__ amax, float* __restrict__ denom) {
  int e = blockIdx.x * blockDim.x + threadIdx.x;
  if (e >= EE) return;
  int row = (int)ei[e];
#pragma unroll
  for (int h = 0; h < H; ++h) {
    float a = expf(alpha[(size_t)e * H + h] - amax[(size_t)row * H + h]);
    alpha[(size_t)e * H + h] = a;
    atomicAdd(&denom[(size_t)row * H + h], a);
  }
}

// normalize alpha in place, then scatter alpha * xw[row] into edge_feat[col]
__global__ void edge_agg_k(const long long* __restrict__ ei, float* __restrict__ alpha,
                           const float* __restrict__ denom, const float* __restrict__ xw,
                           float* __restrict__ efeat) {
  int e = blockIdx.x * blockDim.x + threadIdx.x;
  if (e >= EE) return;
  int row = (int)ei[e];
  int col = (int)ei[(size_t)EE + e];
  const float* xr = xw + (size_t)row * C1;
  float* ef = efeat + (size_t)col * C1;
#pragma unroll
  for (int h = 0; h < H; ++h) {
    float w = alpha[(size_t)e * H + h] / denom[(size_t)row * H + h];
    alpha[(size_t)e * H + h] = w;
#pragma unroll
    for (int c = 0; c < O1; ++c)
      atomicAdd(&ef[h * O1 + c], w * xr[h * O1 + c]);
  }
}

// scatter alpha * edge_feat[col] into hbuf[row]
__global__ void node_agg_k(const long long* __restrict__ ei, const float* __restrict__ alpha,
                           const float* __restrict__ efeat, float* __restrict__ hbuf) {
  int e = blockIdx.x * blockDim.x + threadIdx.x;
  if (e >= EE) return;
  int row = (int)ei[e];
  int col = (int)ei[(size_t)EE + e];
  const float* ef = efeat + (size_t)col * C1;
  float* hb = hbuf + (size_t)row * C1;
#pragma unroll
  for (int h = 0; h < H; ++h) {
    float w = alpha[(size_t)e * H + h];
#pragma unroll
    for (int c = 0; c < O1; ++c)
      atomicAdd(&hb[h * O1 + c], w * ef[h * O1 + c]);
  }
}

// h = elu(Dinv * hbuf + b1), in place
__global__ void finish_h_k(float* __restrict__ hbuf, const float* __restrict__ Dinv,
                           const float* __restrict__ b1) {
  long i = (long)blockIdx.x * blockDim.x + threadIdx.x;
  if (i >= (long)NN * C1) return;
  float v = Dinv[i >> 6] * hbuf[i] + b1[i & 63];
  hbuf[i] = (v > 0.0f) ? v : expm1f(v);
}

__global__ void edge_agg2_k(const long long* __restrict__ ei, const float* __restrict__ xw2,
                            float* __restrict__ efeat2) {
  int e = blockIdx.x * blockDim.x + threadIdx.x;
  if (e >= EE) return;
  int row = (int)ei[e];
  int col = (int)ei[(size_t)EE + e];
#pragma unroll
  for (int j = 0; j < O2; ++j)
    atomicAdd(&efeat2[(size_t)col * O2 + j], xw2[(size_t)row * O2 + j]);
}

__global__ void node_agg2_k(const long long* __restrict__ ei, const float* __restrict__ efeat2,
                            float* __restrict__ out2) {
  int e = blockIdx.x * blockDim.x + threadIdx.x;
  if (e >= EE) return;
  int row = (int)ei[e];
  int col = (int)ei[(size_t)EE + e];
#pragma unroll
  for (int j = 0; j < O2; ++j)
    atomicAdd(&out2[(size_t)row * O2 + j], efeat2[(size_t)col * O2 + j]);
}

// out = log_softmax(Dinv*out2 + b2, axis=1)
__global__ void finalize_k(const float* __restrict__ out2, const float* __restrict__ Dinv,
                           const float* __restrict__ b2, float* __restrict__ out) {
  int n = blockIdx.x * blockDim.x + threadIdx.x;
  if (n >= NN) return;
  float d = Dinv[n];
  float o[O2];
  float mx = -__builtin_inff();
#pragma unroll
  for (int j = 0; j < O2; ++j) {
    o[j] = d * out2[(size_t)n * O2 + j] + b2[j];
    mx = fmaxf(mx, o[j]);
  }
  float s = 0.0f;
#pragma unroll
  for (int j = 0; j < O2; ++j) s += expf(o[j] - mx);
  float lse = mx + logf(s);
#pragma unroll
  for (int j = 0; j < O2; ++j) out[(size_t)n * O2 + j] = o[j] - lse;
}

// ---------------- launcher ----------------
static inline int cdiv(long a, int b) { return (int)((a + b - 1) / b); }

extern "C" void kernel_launch(void* const* d_in, const int* in_sizes, int n_in,
                              void* d_out, int out_size, void* d_ws, size_t ws_size,
                              hipStream_t stream) {
  const float*     x   = (const float*)d_in[0];
  const long long* ei  = (const long long*)d_in[1];   // int64 edge_index [2,E]
  const float*     W1  = (const float*)d_in[2];
  const float*     att = (const float*)d_in[3];
  const float*     b1  = (const float*)d_in[4];
  const float*     W2  = (const float*)d_in[5];
  const float*     b2  = (const float*)d_in[6];
  float*           out = (float*)d_out;

  // workspace layout (floats); total ~36.2M floats (~145 MB)
  float* ws     = (float*)d_ws;
  float* xw1    = ws;                                  // N*64
  float* amax   = xw1    + (size_t)NN * C1;            // N*8
  float* denom  = amax   + (size_t)NN * H;             // N*8
  float* Dinv   = denom  + (size_t)NN * H;             // N
  float* Binv   = Dinv   + NN;                         // M
  float* alpha  = Binv   + MM;                         // E*8
  float* efeat1 = alpha  + (size_t)EE * H;             // M*64
  float* hbuf   = efeat1 + (size_t)MM * C1;            // N*64
  float* xw2    = hbuf   + (size_t)NN * C1;            // N*8 (fully overwritten)
  float* efeat2 = xw2    + (size_t)NN * O2;            // M*8
  float* out2   = efeat2 + (size_t)MM * O2;            // N*8

  const int B = 256;

  // zero-init accumulators (contiguous runs); amax needs -inf
  hipMemsetAsync(denom,  0, ((size_t)NN * H + NN + MM) * sizeof(float), stream);
  hipMemsetAsync(efeat1, 0, ((size_t)MM * C1 + (size_t)NN * C1) * sizeof(float), stream);
  hipMemsetAsync(efeat2, 0, ((size_t)MM * O2 + (size_t)NN * O2) * sizeof(float), stream);
  fill_k<<<cdiv((long)NN * H, B), B, 0, stream>>>(amax, -__builtin_inff(), (long)NN * H);

  // degrees -> inverse degrees (Dinv and Binv are contiguous)
  count_deg_k<<<cdiv(EE, B), B, 0, stream>>>(ei, Dinv, Binv);
  invert_k<<<cdiv((long)NN + MM, B), B, 0, stream>>>(Dinv, (long)NN + MM);

  // hconv1: xw1 = x @ W1 (fp32 WMMA, TDM-staged LDS pipeline)
  gemm1_wmma<<<cdiv(NN, 128), B, 0, stream>>>(x, W1, xw1);

  // attention softmax over incidences grouped by row
  alpha_max_k<<<cdiv(EE, B), B, 0, stream>>>(ei, xw1, att, alpha, amax);
  exp_sum_k  <<<cdiv(EE, B), B, 0, stream>>>(ei, alpha, amax, denom);

  // node -> hyperedge, scale by Binv
  edge_agg_k<<<cdiv(EE, B), B, 0, stream>>>(ei, alpha, denom, xw1, efeat1);
  scale_rows_k<<<cdiv((long)MM * C1, B), B, 0, stream>>>(efeat1, Binv, 6, (long)MM * C1);

  // hyperedge -> node, scale by Dinv, +b1, elu
  node_agg_k<<<cdiv(EE, B), B, 0, stream>>>(ei, alpha, efeat1, hbuf);
  finish_h_k<<<cdiv((long)NN * C1, B), B, 0, stream>>>(hbuf, Dinv, b1);

  // hconv2: xw2 = h @ W2 (fp32 WMMA, padded tile)
  gemm2_wmma<<<cdiv((long)(NN / 16) * 32, B), B, 0, stream>>>(hbuf, W2, xw2);

  edge_agg2_k<<<cdiv(EE, B), B, 0, stream>>>(ei, xw2, efeat2);
  scale_rows_k<<<cdiv((long)MM * O2, B), B, 0, stream>>>(efeat2, Binv, 3, (long)MM * O2);
  node_agg2_k<<<cdiv(EE, B), B, 0, stream>>>(ei, efeat2, out2);

  finalize_k<<<cdiv(NN, B), B, 0, stream>>>(out2, Dinv, b2, out);
}